// GATLayer_59219009077969
// MI455X (gfx1250) — compile-verified
//
#include <hip/hip_runtime.h>
#include <hip/hip_bf16.h>
#include <cstdint>

#define LRELU_ALPHA 0.2f
#define F_IN 128
#define F_OUT 32

typedef __attribute__((ext_vector_type(2))) float v2f;
typedef __attribute__((ext_vector_type(8))) float v8f;

// ---------------------------------------------------------------------------
// Kernel 1: initialize accumulators (h=0 in d_out, m=-inf, s=0)
// ---------------------------------------------------------------------------
__global__ void gat_init(float* __restrict__ h, float* __restrict__ m,
                         float* __restrict__ s, int N) {
    int idx = blockIdx.x * blockDim.x + threadIdx.x;
    if (idx < N) { m[idx] = -INFINITY; s[idx] = 0.f; }
    if (idx < N * F_OUT) h[idx] = 0.f;
}

// ---------------------------------------------------------------------------
// Kernel 2: Wx = x @ W via V_WMMA_F32_16X16X4_F32.
// One wave computes a 16(row)x16(col) tile of Wx; 2 waves/block cover 32 cols.
// A 16x4 f32 fragment: lanes 0-15 hold {K=k,K=k+1} for M=lane,
//                      lanes 16-31 hold {K=k+2,K=k+3} for M=lane-16.
// B 4x16 f32 fragment mirrors A with lanes meaning N.
// C/D 16x16 f32: VGPR r -> M=r (lanes 0-15) / M=r+8 (lanes 16-31), N=lane%16.
// ---------------------------------------------------------------------------
__global__ void __launch_bounds__(64)
gat_gemm_wmma(const float* __restrict__ x, const float* __restrict__ W,
              float* __restrict__ Wx, int N) {
    const int wave = threadIdx.x >> 5;   // 0..1 -> column tile
    const int lane = threadIdx.x & 31;
    const int half = lane >> 4;          // 0 or 1
    const int l16  = lane & 15;
    const int row0 = blockIdx.x * 16;
    const int col0 = wave * 16;

    // clamp row for A-loads so EXEC stays all-ones on a ragged tail
    const int arow = min(row0 + l16, N - 1);
    const float* xrow = x + (size_t)arow * F_IN;

    v8f c = {};
#pragma unroll
    for (int k = 0; k < F_IN; k += 4) {
        float2 av = *(const float2*)(xrow + k + 2 * half);
        v2f a; a.x = av.x; a.y = av.y;
        v2f b;
        b.x = W[(k + 2 * half)     * F_OUT + col0 + l16];
        b.y = W[(k + 2 * half + 1) * F_OUT + col0 + l16];
        c = __builtin_amdgcn_wmma_f32_16x16x4_f32(
                /*neg_a=*/false, a, /*neg_b=*/false, b,
                /*c_mod=*/(short)0, c, /*reuse_a=*/false, /*reuse_b=*/false);
    }
#pragma unroll
    for (int r = 0; r < 8; ++r) {
        int row = row0 + r + 8 * half;
        if (row < N)
            Wx[(size_t)row * F_OUT + col0 + l16] = c[r];
    }
}

// ---------------------------------------------------------------------------
// Kernel 3: per-node attention dots  alpha_src[n]=Wx[n]·a[:32], alpha_dst=·a[32:]
// ---------------------------------------------------------------------------
__global__ void gat_alpha(const float* __restrict__ Wx, const float* __restrict__ a,
                          float* __restrict__ asrc, float* __restrict__ adst, int N) {
    int n = blockIdx.x * blockDim.x + threadIdx.x;
    if (n >= N) return;
    const float* row = Wx + (size_t)n * F_OUT;
    float s1 = 0.f, s2 = 0.f;
#pragma unroll
    for (int j = 0; j < F_OUT; ++j) {
        float v = row[j];
        s1 += v * a[j];
        s2 += v * a[F_OUT + j];
    }
    asrc[n] = s1;
    adst[n] = s2;
}

// ---------------------------------------------------------------------------
// Kernel 4: edge logits + LeakyReLU + segment max (hw float atomic max)
// ---------------------------------------------------------------------------
__global__ void gat_edge_logits(const long long* __restrict__ src,
                                const long long* __restrict__ dst,
                                const float* __restrict__ asrc,
                                const float* __restrict__ adst,
                                float* __restrict__ e_ws, float* __restrict__ m,
                                int E) {
    int i = blockIdx.x * blockDim.x + threadIdx.x;
    if (i >= E) return;
    int s = (int)src[i], d = (int)dst[i];
    float e = asrc[s] + adst[d];
    e = (e >= 0.f) ? e : LRELU_ALPHA * e;
    e_ws[i] = e;
    __hip_atomic_fetch_max(&m[s], e, __ATOMIC_RELAXED, __HIP_MEMORY_SCOPE_AGENT);
}

// ---------------------------------------------------------------------------
// Kernel 5: exp(e - m[src]) + segment sum (hw float atomic add)
// ---------------------------------------------------------------------------
__global__ void gat_edge_exp(const long long* __restrict__ src,
                             float* __restrict__ e_ws,
                             const float* __restrict__ m,
                             float* __restrict__ ssum, int E) {
    int i = blockIdx.x * blockDim.x + threadIdx.x;
    if (i >= E) return;
    int s = (int)src[i];
    float ev = __expf(e_ws[i] - m[s]);   // m[src] always finite for a live edge
    e_ws[i] = ev;
    __hip_atomic_fetch_add(&ssum[s], ev, __ATOMIC_RELAXED, __HIP_MEMORY_SCOPE_AGENT);
}

// ---------------------------------------------------------------------------
// Kernel 6: aggregation. One lane per (edge, feature): each wave does one
// coalesced 128B gather of Wx[dst] and one coalesced 128B atomic-add burst
// into h[src]. ssum[src] > 0 for every live edge, so no ==0 guard needed.
// ---------------------------------------------------------------------------
__global__ void gat_aggregate(const long long* __restrict__ src,
                              const long long* __restrict__ dst,
                              const float* __restrict__ e_ws,
                              const float* __restrict__ ssum,
                              const float* __restrict__ Wx,
                              float* __restrict__ h, int E) {
    long long tid = (long long)blockIdx.x * blockDim.x + threadIdx.x;
    int i = (int)(tid >> 5);
    int f = (int)(tid & 31);
    if (i >= E) return;
    int s = (int)src[i], d = (int)dst[i];
    float att = e_ws[i] / ssum[s];
    float v = att * Wx[(size_t)d * F_OUT + f];
    __hip_atomic_fetch_add(&h[(size_t)s * F_OUT + f], v,
                           __ATOMIC_RELAXED, __HIP_MEMORY_SCOPE_AGENT);
}

// ---------------------------------------------------------------------------
// Kernel 7: ELU in place on d_out
// ---------------------------------------------------------------------------
__global__ void gat_elu(float* __restrict__ h, int n) {
    int idx = blockIdx.x * blockDim.x + threadIdx.x;
    if (idx >= n) return;
    float v = h[idx];
    h[idx] = (v > 0.f) ? v : expm1f(v);
}

// ---------------------------------------------------------------------------
extern "C" void kernel_launch(void* const* d_in, const int* in_sizes, int n_in,
                              void* d_out, int out_size, void* d_ws, size_t ws_size,
                              hipStream_t stream) {
    const float*     x  = (const float*)d_in[0];      // (N, 128)
    const float*     W  = (const float*)d_in[1];      // (128, 32)
    const float*     a  = (const float*)d_in[2];      // (64, 1)
    const long long* ei = (const long long*)d_in[3];  // (2, E) int64

    const int N = in_sizes[0] / F_IN;
    const int E = in_sizes[3] / 2;
    const long long* src = ei;
    const long long* dst = ei + E;

    // workspace layout (floats)
    float* Wx   = (float*)d_ws;                 // N*32
    float* asrc = Wx + (size_t)N * F_OUT;       // N
    float* adst = asrc + N;                     // N
    float* m    = adst + N;                     // N
    float* ssum = m + N;                        // N
    float* e_ws = ssum + N;                     // E

    float* h = (float*)d_out;                   // N*32, accumulate then ELU

    const int T = 256;
    gat_init<<<(N * F_OUT + T - 1) / T, T, 0, stream>>>(h, m, ssum, N);

    gat_gemm_wmma<<<(N + 15) / 16, 64, 0, stream>>>(x, W, Wx, N);

    gat_alpha<<<(N + T - 1) / T, T, 0, stream>>>(Wx, a, asrc, adst, N);

    gat_edge_logits<<<(E + T - 1) / T, T, 0, stream>>>(src, dst, asrc, adst, e_ws, m, E);

    gat_edge_exp<<<(E + T - 1) / T, T, 0, stream>>>(src, e_ws, m, ssum, E);

    long long agg_threads = (long long)E * 32;
    gat_aggregate<<<(int)((agg_threads + T - 1) / T), T, 0, stream>>>(
        src, dst, e_ws, ssum, Wx, h, E);

    gat_elu<<<(N * F_OUT + T - 1) / T, T, 0, stream>>>(h, N * F_OUT);
}